// NanoGPTModel_29094108463549
// MI455X (gfx1250) — compile-verified
//
#include <hip/hip_runtime.h>
#include <math.h>

typedef __attribute__((ext_vector_type(2))) float v2f;
typedef __attribute__((ext_vector_type(8))) float v8f;

// ---------------- model dims ----------------
constexpr int Vv  = 50257;
constexpr int Dd  = 1024;
constexpr int Tt  = 1024;
constexpr int Hh  = 16;
constexpr int Ll  = 6;
constexpr int Bb  = 2;
constexpr int HSs = 64;           // head size
constexpr int Mrows = Bb * Tt;    // 2048 token rows

// ---------------- gemm flags ----------------
enum { GF_BIAS = 1, GF_RELU = 2, GF_RES = 4, GF_HEADW = 8, GF_NT = 16 };

// ============================================================
// Embedding: x[r, d] = tok_emb[idx[r], d] + pos_emb[r % T, d]
// ============================================================
__global__ __launch_bounds__(256) void embed_kernel(
    const int* __restrict__ idx, const float* __restrict__ tok,
    const float* __restrict__ pos, float* __restrict__ x)
{
    int r = blockIdx.x;
    int t = r % Tt;
    int tokid = idx[r];
    const float* te = tok + (size_t)tokid * Dd;
    const float* pe = pos + (size_t)t * Dd;
    float* xr = x + (size_t)r * Dd;
    for (int d = threadIdx.x; d < Dd; d += 256)
        xr[d] = te[d] + pe[d];
}

// ============================================================
// LayerNorm over D=1024, one block (256 threads) per row
// ============================================================
__global__ __launch_bounds__(256) void ln_kernel(
    const float* __restrict__ x, const float* __restrict__ g,
    const float* __restrict__ beta, float* __restrict__ out)
{
    int r = blockIdx.x, tid = threadIdx.x;
    const float* row = x + (size_t)r * Dd;
    float s = 0.f, s2 = 0.f;
    for (int d = tid; d < Dd; d += 256) { float v = row[d]; s += v; s2 += v * v; }
    __shared__ float rs[256], rs2[256];
    rs[tid] = s; rs2[tid] = s2; __syncthreads();
    for (int off = 128; off > 0; off >>= 1) {
        if (tid < off) { rs[tid] += rs[tid + off]; rs2[tid] += rs2[tid + off]; }
        __syncthreads();
    }
    float mu  = rs[0] * (1.f / Dd);
    float var = rs2[0] * (1.f / Dd) - mu * mu;
    float inv = rsqrtf(var + 1e-5f);
    float* orow = out + (size_t)r * Dd;
    for (int d = tid; d < Dd; d += 256)
        orow[d] = (row[d] - mu) * inv * g[d] + beta[d];
}

// ============================================================
// FP32 WMMA GEMM:  C[M,N] = op( A[M,K] @ B[K,N] (+bias) (+resid) )
// One wave computes one 16x16 tile via V_WMMA_F32_16X16X4_F32.
// Block = 128 threads = 4 waves -> 16x64 output per block.
// GF_HEADW: B layout is (H, K, 64) — per-64-col group uses its own
// K x 64 slab (valid since 16 | 64).
// Full tiles take a branch-free hot loop (no EXEC churn); ragged
// N-edge tiles (lm head only) use clamped columns + 0/1 mask.
// ============================================================
__global__ __launch_bounds__(128) void gemm_wmma(
    const float* __restrict__ A, int lda,
    const float* __restrict__ Bm, int ldb,
    float* __restrict__ C, int ldc,
    const float* __restrict__ bias,
    const float* __restrict__ resid, int ldr,
    int M, int N, int K, int flags)
{
    int wave = threadIdx.x >> 5;
    int lane = threadIdx.x & 31;
    int row_base = blockIdx.y * 16;
    int col_base = blockIdx.x * 64 + wave * 16;
    if (col_base >= N) return;                 // wave-uniform: EXEC stays full

    const float* Bp = Bm;
    int ldbe = ldb;
    int cb   = col_base;
    if (flags & GF_HEADW) {
        Bp   = Bm + (size_t)(col_base >> 6) * ((size_t)K * 64);
        ldbe = 64;
        cb   = col_base & 63;
    }

    int  lh   = lane & 15;
    int  koff = (lane >> 4) << 1;              // lanes 0-15 -> K{0,1}, 16-31 -> K{2,3}
    int  arow = row_base + lh;                 // A: lane indexes M
    const float* Aptr = A + (size_t)arow * lda + koff;

    // wave-uniform full-tile predicate: HEADW tiles always full
    bool full = (flags & GF_HEADW) || (col_base + 15 < N);

    v8f acc = {};
    if (full) {
        const float* Bptr = Bp + (size_t)koff * ldbe + (cb + lh);
        for (int k0 = 0; k0 < K; k0 += 4) {
            v2f a, b;
            a.x = Aptr[0];
            a.y = Aptr[1];
            b.x = Bptr[0];
            b.y = Bptr[ldbe];
            acc = __builtin_amdgcn_wmma_f32_16x16x4_f32(
                      false, a, false, b, (short)0, acc, false, false);
            Aptr += 4;
            Bptr += (size_t)4 * ldbe;
        }
    } else {
        int   col   = col_base + lh;
        bool  valid = col < N;
        float msk   = valid ? 1.f : 0.f;
        int   scol  = valid ? col : (N - 1);   // clamped: always in-bounds
        const float* Bptr = Bp + (size_t)koff * ldbe + scol;
        for (int k0 = 0; k0 < K; k0 += 4) {
            v2f a, b;
            a.x = Aptr[0];
            a.y = Aptr[1];
            b.x = Bptr[0]    * msk;            // branchless, no EXEC ops
            b.y = Bptr[ldbe] * msk;
            acc = __builtin_amdgcn_wmma_f32_16x16x4_f32(
                      false, a, false, b, (short)0, acc, false, false);
            Aptr += 4;
            Bptr += (size_t)4 * ldbe;
        }
    }

    int ccol = col_base + lh;
    if (ccol >= N) return;                     // after all WMMAs: per-lane guard ok
    float bv = (flags & GF_BIAS) ? bias[ccol] : 0.f;
    int mrow0 = row_base + ((lane >> 4) << 3); // lanes 16-31 write rows M+8..M+15
#pragma unroll
    for (int i = 0; i < 8; ++i) {
        int mrow = mrow0 + i;
        float val = acc[i] + bv;
        if (flags & GF_RELU) val = fmaxf(val, 0.f);
        if (flags & GF_RES)  val += resid[(size_t)mrow * ldr + ccol];
        if (flags & GF_NT)
            __builtin_nontemporal_store(val, &C[(size_t)mrow * ldc + ccol]);
        else
            C[(size_t)mrow * ldc + ccol] = val;
    }
}

// ============================================================
// Causal attention, one block (256 thr) per (b, h, t) query row.
// q/k/v/o are [B*T, D] with column = h*64 + kk.
// ============================================================
__global__ __launch_bounds__(256) void attn_kernel(
    const float* __restrict__ q, const float* __restrict__ k,
    const float* __restrict__ v, float* __restrict__ o)
{
    int t  = blockIdx.x;
    int bh = blockIdx.y;
    int b  = bh / Hh, h = bh % Hh;
    int tid = threadIdx.x;
    size_t bT = (size_t)b * Tt;
    int    hh = h * HSs;

    __shared__ float sc[Tt];
    __shared__ float red[256];
    __shared__ float qv[HSs];

    const float* qrow = q + (bT + t) * Dd + hh;
    if (tid < HSs) qv[tid] = qrow[tid];
    __syncthreads();

    // scores
    for (int s = tid; s <= t; s += 256) {
        const float* krow = k + (bT + s) * Dd + hh;
        float acc = 0.f;
#pragma unroll
        for (int d = 0; d < HSs; ++d) acc += qv[d] * krow[d];
        sc[s] = acc * 0.125f;                  // 1/sqrt(64)
    }
    __syncthreads();

    // max
    float m = -__builtin_inff();
    for (int s = tid; s <= t; s += 256) m = fmaxf(m, sc[s]);
    red[tid] = m; __syncthreads();
    for (int off = 128; off > 0; off >>= 1) {
        if (tid < off) red[tid] = fmaxf(red[tid], red[tid + off]);
        __syncthreads();
    }
    m = red[0]; __syncthreads();

    // exp + sum
    float ssum = 0.f;
    for (int s = tid; s <= t; s += 256) { float e = expf(sc[s] - m); sc[s] = e; ssum += e; }
    red[tid] = ssum; __syncthreads();
    for (int off = 128; off > 0; off >>= 1) {
        if (tid < off) red[tid] += red[tid + off];
        __syncthreads();
    }
    float inv = 1.f / red[0];
    __syncthreads();                           // everyone has read red[0]

    // o[t, kk] = sum_s p[s] * v[s, kk] ; 4 partials per output dim
    int kk = tid & 63, part = tid >> 6;
    float acc = 0.f;
    for (int s = part; s <= t; s += 4)
        acc += sc[s] * v[(bT + s) * Dd + hh + kk];
    red[tid] = acc; __syncthreads();
    if (part == 0) {
        float r2 = (red[kk] + red[kk + 64]) + (red[kk + 128] + red[kk + 192]);
        o[(bT + t) * Dd + hh + kk] = r2 * inv;
    }
}

// ============================================================
// Per-row NLL: nll[r] = logsumexp(logits[r,:]) - logits[r, target[r]]
// ============================================================
__global__ __launch_bounds__(256) void nll_kernel(
    const float* __restrict__ logits, const int* __restrict__ targets,
    float* __restrict__ nll)
{
    int r = blockIdx.x, tid = threadIdx.x;
    const float* row = logits + (size_t)r * Vv;
    __shared__ float red[256];

    float m = -__builtin_inff();
    for (int j = tid; j < Vv; j += 256) m = fmaxf(m, row[j]);
    red[tid] = m; __syncthreads();
    for (int off = 128; off > 0; off >>= 1) {
        if (tid < off) red[tid] = fmaxf(red[tid], red[tid + off]);
        __syncthreads();
    }
    m = red[0]; __syncthreads();

    float s = 0.f;
    for (int j = tid; j < Vv; j += 256) s += expf(row[j] - m);
    red[tid] = s; __syncthreads();
    for (int off = 128; off > 0; off >>= 1) {
        if (tid < off) red[tid] += red[tid + off];
        __syncthreads();
    }
    if (tid == 0) {
        float lse = m + logf(red[0]);
        nll[r] = lse - row[targets[r]];
    }
}

__global__ __launch_bounds__(256) void loss_reduce(
    const float* __restrict__ nll, float* __restrict__ out, int n)
{
    __shared__ float red[256];
    int tid = threadIdx.x;
    float s = 0.f;
    for (int i = tid; i < n; i += 256) s += nll[i];
    red[tid] = s; __syncthreads();
    for (int off = 128; off > 0; off >>= 1) {
        if (tid < off) red[tid] += red[tid + off];
        __syncthreads();
    }
    if (tid == 0) *out = red[0] / (float)n;
}

// ============================================================
// Orchestration
// ============================================================
extern "C" void kernel_launch(void* const* d_in, const int* in_sizes, int n_in,
                              void* d_out, int out_size, void* d_ws, size_t ws_size,
                              hipStream_t stream) {
    (void)in_sizes; (void)n_in; (void)out_size; (void)ws_size;

    const int*   idx     = (const int*)  d_in[0];
    const int*   targets = (const int*)  d_in[1];
    const float* tok_emb = (const float*)d_in[2];
    const float* pos_emb = (const float*)d_in[3];
    const float* wq      = (const float*)d_in[4];
    const float* wk      = (const float*)d_in[5];
    const float* wv      = (const float*)d_in[6];
    const float* wproj   = (const float*)d_in[7];
    const float* bproj   = (const float*)d_in[8];
    const float* ln1_g   = (const float*)d_in[9];
    const float* ln1_b   = (const float*)d_in[10];
    const float* ln2_g   = (const float*)d_in[11];
    const float* ln2_b   = (const float*)d_in[12];
    const float* w1      = (const float*)d_in[13];
    const float* b1      = (const float*)d_in[14];
    const float* w2      = (const float*)d_in[15];
    const float* b2      = (const float*)d_in[16];
    const float* lnf_g   = (const float*)d_in[17];
    const float* lnf_b   = (const float*)d_in[18];
    const float* lm_w    = (const float*)d_in[19];
    const float* lm_b    = (const float*)d_in[20];
    float* logits = (float*)d_out;

    // workspace layout (MD = 8 MB each); act aliases dead q/k/v/o region
    char*  ws = (char*)d_ws;
    size_t MD = (size_t)Mrows * Dd * sizeof(float);
    float* x    = (float*)(ws);
    float* h    = (float*)(ws + 1 * MD);
    float* q    = (float*)(ws + 2 * MD);
    float* kbuf = (float*)(ws + 3 * MD);
    float* vbuf = (float*)(ws + 4 * MD);
    float* o    = (float*)(ws + 5 * MD);
    float* x2   = (float*)(ws + 6 * MD);
    float* act  = (float*)(ws + 2 * MD);   // 32 MB, reuses q..o after attention
    float* nll  = (float*)(ws + 7 * MD);

    embed_kernel<<<Mrows, 256, 0, stream>>>(idx, tok_emb, pos_emb, x);

    dim3 gD((Dd + 63) / 64, Mrows / 16);        // N = 1024 GEMMs
    dim3 g4D((4 * Dd + 63) / 64, Mrows / 16);   // N = 4096 GEMM
    dim3 gAttn(Tt, Bb * Hh);

    for (int l = 0; l < Ll; ++l) {
        const float* wq_l = wq    + (size_t)l * Hh * Dd * HSs;
        const float* wk_l = wk    + (size_t)l * Hh * Dd * HSs;
        const float* wv_l = wv    + (size_t)l * Hh * Dd * HSs;
        const float* wp_l = wproj + (size_t)l * Dd * Dd;
        const float* bp_l = bproj + (size_t)l * Dd;
        const float* w1_l = w1    + (size_t)l * Dd * 4 * Dd;
        const float* b1_l = b1    + (size_t)l * 4 * Dd;
        const float* w2_l = w2    + (size_t)l * 4 * Dd * Dd;
        const float* b2_l = b2    + (size_t)l * Dd;

        // h = LN1(x)
        ln_kernel<<<Mrows, 256, 0, stream>>>(x, ln1_g + (size_t)l * Dd,
                                             ln1_b + (size_t)l * Dd, h);
        // q,k,v = h @ W{q,k,v}  (per-head weight layout)
        gemm_wmma<<<gD, 128, 0, stream>>>(h, Dd, wq_l, 0, q,    Dd, nullptr, nullptr, 0,
                                          Mrows, Dd, Dd, GF_HEADW);
        gemm_wmma<<<gD, 128, 0, stream>>>(h, Dd, wk_l, 0, kbuf, Dd, nullptr, nullptr, 0,
                                          Mrows, Dd, Dd, GF_HEADW);
        gemm_wmma<<<gD, 128, 0, stream>>>(h, Dd, wv_l, 0, vbuf, Dd, nullptr, nullptr, 0,
                                          Mrows, Dd, Dd, GF_HEADW);
        // o = softmax(q k^T / sqrt(HS), causal) v   (heads concatenated)
        attn_kernel<<<gAttn, 256, 0, stream>>>(q, kbuf, vbuf, o);
        // x2 = x + o @ Wp + bp
        gemm_wmma<<<gD, 128, 0, stream>>>(o, Dd, wp_l, Dd, x2, Dd, bp_l, x, Dd,
                                          Mrows, Dd, Dd, GF_BIAS | GF_RES);
        // h = LN2(x2)
        ln_kernel<<<Mrows, 256, 0, stream>>>(x2, ln2_g + (size_t)l * Dd,
                                             ln2_b + (size_t)l * Dd, h);
        // act = relu(h @ W1 + b1)
        gemm_wmma<<<g4D, 128, 0, stream>>>(h, Dd, w1_l, 4 * Dd, act, 4 * Dd, b1_l,
                                           nullptr, 0, Mrows, 4 * Dd, Dd,
                                           GF_BIAS | GF_RELU);
        // x = x2 + act @ W2 + b2
        gemm_wmma<<<gD, 128, 0, stream>>>(act, 4 * Dd, w2_l, Dd, x, Dd, b2_l, x2, Dd,
                                          Mrows, Dd, 4 * Dd, GF_BIAS | GF_RES);
    }

    // h = LNf(x)
    ln_kernel<<<Mrows, 256, 0, stream>>>(x, lnf_g, lnf_b, h);

    // logits = h @ lm_w + lm_b   (N = 50257, guarded; NT stores — logits
    // are written once and exceed L2, keep L2 for lm_w reuse)
    dim3 gV((Vv + 63) / 64, Mrows / 16);
    gemm_wmma<<<gV, 128, 0, stream>>>(h, Dd, lm_w, Vv, logits, Vv, lm_b, nullptr, 0,
                                      Mrows, Vv, Dd, GF_BIAS | GF_NT);

    // loss
    nll_kernel<<<Mrows, 256, 0, stream>>>(logits, targets, nll);
    loss_reduce<<<1, 256, 0, stream>>>(nll, logits + (size_t)Mrows * Vv, Mrows);
}